// SkipGCN_51616916963797
// MI455X (gfx1250) — compile-verified
//
#include <hip/hip_runtime.h>
#include <math.h>

#define IN_C 512
#define HID 100
#define OUT_C 2
#define NT 8          // 7 tiles cover HID=100 (pad 112) + 1 tile for the 2 skip cols
#define WP_FLOATS (256 * NT * 16 * 2)   // kp pairs x tiles x cols x 2 = 65536 floats

typedef __attribute__((ext_vector_type(2))) float v2f;
typedef __attribute__((ext_vector_type(8))) float v8f;

// ---------------- degree / norm ----------------
__global__ void deg_init_kernel(float* deg, int n) {
  int i = blockIdx.x * blockDim.x + threadIdx.x;
  if (i < n) deg[i] = 1.0f;  // self-loop contributes 1 to every node's degree
}

__global__ void deg_accum_kernel(const int* __restrict__ dst, float* deg, int e) {
  int i = blockIdx.x * blockDim.x + threadIdx.x;
  if (i < e) atomicAdd(&deg[dst[i]], 1.0f);
}

__global__ void deg_rsqrt_kernel(float* deg, int n) {
  int i = blockIdx.x * blockDim.x + threadIdx.x;
  if (i < n) { float d = deg[i]; deg[i] = (d > 0.f) ? rsqrtf(d) : 0.f; }
}

// ---------------- pack W1 (512x100) + Ws (512x2) into padded interleaved Wp ----
// Wp[kp][t][col][kin] with kp = k/2, kin = k%2; tiles 0..6 = W1 cols t*16+col
// (zero past col 99), tile 7 = Ws (cols 0..1, rest zero). One b64 load per frag.
__global__ void pack_w_kernel(const float* __restrict__ W1,
                              const float* __restrict__ Ws,
                              float* __restrict__ Wp) {
  int idx = blockIdx.x * blockDim.x + threadIdx.x;
  if (idx >= WP_FLOATS) return;
  int kin = idx & 1;
  int col = (idx >> 1) & 15;
  int t   = (idx >> 5) & 7;
  int kp  = idx >> 8;
  int k   = (kp << 1) + kin;
  float v = 0.f;
  if (t < 7) {
    int c = (t << 4) + col;
    if (c < HID) v = W1[(size_t)k * HID + c];
  } else {
    if (col < OUT_C) v = Ws[(size_t)k * OUT_C + col];
  }
  Wp[idx] = v;
}

// ---------------- fused GEMM: h = x@W1, skip = x@Ws, agg = h*dinv^2 ------------
// One wave per 16-row tile. Branch-free inner loop: 1 A b64-load + 8 B b64-loads
// (constant offsets, single clause) + 8 x V_WMMA_F32_16X16X4_F32 per K-step.
__global__ void __launch_bounds__(256)
gemm1_wmma_kernel(const float* __restrict__ x, const float* __restrict__ Wp,
                  const float* __restrict__ dinv, float* __restrict__ h,
                  float* __restrict__ agg, float* __restrict__ outp, int n_nodes) {
  const int lane = threadIdx.x & 31;
  const int wv   = threadIdx.x >> 5;
  const int tile = blockIdx.x * (blockDim.x >> 5) + wv;
  const int row0 = tile << 4;
  if (row0 >= n_nodes) return;

  const int m  = lane & 15;         // M index (A) / N index (B,C,D)
  const int kh = (lane >> 4) << 1;  // lane-half selects K pair {0,1} vs {2,3}

  v8f acc[NT] = {};

  const float* __restrict__ pa = x + (size_t)(row0 + m) * IN_C + kh;
  const float* __restrict__ pb = Wp + ((size_t)(lane >> 4) * 256) + (m << 1);

  for (int k0 = 0; k0 < IN_C; k0 += 4) {
    v2f a = *(const v2f*)pa;               // A frag: K=kh,kh+1 within this step
#pragma unroll
    for (int t = 0; t < NT; ++t) {
      v2f b = *(const v2f*)(pb + (t << 5)); // B frag for tile t, zero-padded
      acc[t] = __builtin_amdgcn_wmma_f32_16x16x4_f32(
          false, a, false, b, (short)0, acc[t], false, false);
    }
    pa += 4;
    pb += 512;                              // advance 2 kp rows (4 K)
  }

  // C/D layout: VGPR r holds M=r (lanes 0-15) and M=8+r (lanes 16-31), N=lane&15
  const int rbase = (lane < 16) ? 0 : 8;
#pragma unroll
  for (int r = 0; r < 8; ++r) {
    const int row = row0 + rbase + r;
    const float di  = dinv[row];
    const float di2 = di * di;
    float* __restrict__ hrow = h   + (size_t)row * HID;
    float* __restrict__ arow = agg + (size_t)row * HID;
#pragma unroll
    for (int t = 0; t < 7; ++t) {
      int col = (t << 4) + m;
      if (col < HID) {
        float v = acc[t][r];
        hrow[col] = v;            // for edge gathers
        arow[col] = v * di2;      // self-loop contribution seeds aggregation
      }
    }
    if (m < OUT_C) outp[(size_t)row * OUT_C + m] = acc[7][r];  // skip = x@Ws
  }
}

// ---------------- edge scatter, stage 1 (wave per edge, 100 feats) ----------------
__global__ void scatter1_kernel(const float* __restrict__ h,
                                const float* __restrict__ dinv,
                                const int* __restrict__ src,
                                const int* __restrict__ dst,
                                float* __restrict__ agg, int n_edges) {
  int lane = threadIdx.x & 31;
  int wv   = threadIdx.x >> 5;
  long long e = (long long)blockIdx.x * (blockDim.x >> 5) + wv;
  if (e >= n_edges) return;
  int s = src[e], d = dst[e];
  float norm = dinv[s] * dinv[d];
  const float* __restrict__ hs = h + (size_t)s * HID;
  float* __restrict__ ad = agg + (size_t)d * HID;
#pragma unroll
  for (int i = 0; i < 4; ++i) {
    int f = lane + (i << 5);
    if (f < HID) atomicAdd(&ad[f], hs[f] * norm);
  }
}

// ---------------- fused: h1=relu(agg+b1); h2=h1@W2; out += b2+bs+selfloop -------
__global__ void gemm2_fused_kernel(const float* __restrict__ agg,
                                   const float* __restrict__ b1,
                                   const float* __restrict__ W2,
                                   const float* __restrict__ b2,
                                   const float* __restrict__ bs,
                                   const float* __restrict__ dinv,
                                   float* __restrict__ h2,
                                   float* __restrict__ out, int n_nodes) {
  int n = blockIdx.x * blockDim.x + threadIdx.x;
  if (n >= n_nodes) return;
  const float* __restrict__ r = agg + (size_t)n * HID;
  float a0 = 0.f, a1 = 0.f;
#pragma unroll 4
  for (int k = 0; k < HID; ++k) {
    float v = r[k] + b1[k];
    v = v > 0.f ? v : 0.f;                 // relu(agg + b1), fused
    a0 += v * W2[k * 2 + 0];
    a1 += v * W2[k * 2 + 1];
  }
  h2[(size_t)n * 2 + 0] = a0;
  h2[(size_t)n * 2 + 1] = a1;
  float di = dinv[n];
  float di2 = di * di;
  out[(size_t)n * 2 + 0] += b2[0] + bs[0] + a0 * di2;   // self-loop + biases
  out[(size_t)n * 2 + 1] += b2[1] + bs[1] + a1 * di2;
}

// ---------------- edge scatter, stage 2 (thread per edge, 2 feats) ----------------
__global__ void scatter2_kernel(const float* __restrict__ h2,
                                const float* __restrict__ dinv,
                                const int* __restrict__ src,
                                const int* __restrict__ dst,
                                float* __restrict__ out, int n_edges) {
  int e = blockIdx.x * blockDim.x + threadIdx.x;
  if (e >= n_edges) return;
  int s = src[e], d = dst[e];
  float norm = dinv[s] * dinv[d];
  atomicAdd(&out[(size_t)d * 2 + 0], h2[(size_t)s * 2 + 0] * norm);
  atomicAdd(&out[(size_t)d * 2 + 1], h2[(size_t)s * 2 + 1] * norm);
}

// ---------------- 2-class log_softmax in place ----------------
__global__ void log_softmax_kernel(float* __restrict__ out, int n_nodes) {
  int n = blockIdx.x * blockDim.x + threadIdx.x;
  if (n >= n_nodes) return;
  float z0 = out[(size_t)n * 2 + 0];
  float z1 = out[(size_t)n * 2 + 1];
  float mx = fmaxf(z0, z1);
  float lse = mx + logf(expf(z0 - mx) + expf(z1 - mx));
  out[(size_t)n * 2 + 0] = z0 - lse;
  out[(size_t)n * 2 + 1] = z1 - lse;
}

extern "C" void kernel_launch(void* const* d_in, const int* in_sizes, int n_in,
                              void* d_out, int out_size, void* d_ws, size_t ws_size,
                              hipStream_t stream) {
  const float* x  = (const float*)d_in[0];
  const int*   ei = (const int*)d_in[1];
  const float* W1 = (const float*)d_in[2];
  const float* b1 = (const float*)d_in[3];
  const float* W2 = (const float*)d_in[4];
  const float* b2 = (const float*)d_in[5];
  const float* Ws = (const float*)d_in[6];
  const float* bs = (const float*)d_in[7];
  float* out = (float*)d_out;

  const int n_nodes = in_sizes[0] / IN_C;
  const int n_edges = in_sizes[1] / 2;
  const int* srcv = ei;            // edge_index row 0
  const int* dstv = ei + n_edges;  // edge_index row 1

  // workspace: Wp[64K] | dinv[N] | h[N*HID] | agg[N*HID] | h2[N*2]  (~81.5 MB)
  float* Wp   = (float*)d_ws;
  float* dinv = Wp + WP_FLOATS;
  size_t off  = ((size_t)n_nodes + 255) & ~(size_t)255;
  float* h    = dinv + off;
  float* agg  = h + (size_t)n_nodes * HID;
  float* h2   = agg + (size_t)n_nodes * HID;

  const int T = 256;

  deg_init_kernel<<<(n_nodes + T - 1) / T, T, 0, stream>>>(dinv, n_nodes);
  deg_accum_kernel<<<(n_edges + T - 1) / T, T, 0, stream>>>(dstv, dinv, n_edges);
  deg_rsqrt_kernel<<<(n_nodes + T - 1) / T, T, 0, stream>>>(dinv, n_nodes);

  pack_w_kernel<<<WP_FLOATS / T, T, 0, stream>>>(W1, Ws, Wp);

  int tiles = (n_nodes + 15) / 16;  // 6250 waves, 8 waves/block
  gemm1_wmma_kernel<<<(tiles + 7) / 8, T, 0, stream>>>(x, Wp, dinv, h, agg, out, n_nodes);

  scatter1_kernel<<<(n_edges + 7) / 8, T, 0, stream>>>(h, dinv, srcv, dstv, agg, n_edges);

  gemm2_fused_kernel<<<(n_nodes + T - 1) / T, T, 0, stream>>>(agg, b1, W2, b2, bs, dinv,
                                                              h2, out, n_nodes);
  scatter2_kernel<<<(n_edges + T - 1) / T, T, 0, stream>>>(h2, dinv, srcv, dstv, out, n_edges);

  log_softmax_kernel<<<(n_nodes + T - 1) / T, T, 0, stream>>>(out, n_nodes);
}